// Model_59803124629558
// MI455X (gfx1250) — compile-verified
//
#include <hip/hip_runtime.h>

// Problem dims (fixed by the reference)
#define B_  64
#define C_  16
#define LM_ 40
#define LE_ 32
#define D_  768
#define R_  2048
#define P_  49
#define M_  4
#define NL_ 2

#define D4_ (D_ / 4)
#define R4_ (R_ / 4)

typedef __attribute__((ext_vector_type(2))) float v2f;
typedef __attribute__((ext_vector_type(8))) float v8f;

__device__ __forceinline__ float dot4(float4 a, float4 b) {
  return a.x * b.x + a.y * b.y + a.z * b.z + a.w * b.w;
}

// ---------------------------------------------------------------------------
// Block-wide sum reduction (blockDim.x == 256)
// ---------------------------------------------------------------------------
__device__ __forceinline__ float block_sum(float v, float* sm) {
  const int t = threadIdx.x;
  sm[t] = v;
  __syncthreads();
  for (int s = 128; s > 0; s >>= 1) {
    if (t < s) sm[t] += sm[t + s];
    __syncthreads();
  }
  const float r = sm[0];
  __syncthreads();
  return r;
}

// ---------------------------------------------------------------------------
// WMMA GEMM: Out[M,N] = A[M,K] @ W[K,N] + bias[N]   (all row-major f32)
// One wave -> 16 (M) x 64 (N) strip: 4 accumulators reuse one A fragment.
// blockDim = 64 (2 waves stacked in M). Requires M%16==0, K%4==0, N%64==0.
// ---------------------------------------------------------------------------
__global__ __launch_bounds__(64)
void gemm16_wmma(const float* __restrict__ A, const float* __restrict__ W,
                 const float* __restrict__ bias, float* __restrict__ Out,
                 int Mrows, int K, int N) {
  const int lane = threadIdx.x & 31;
  const int wave = threadIdx.x >> 5;
  const int n0 = blockIdx.x * 64;
  const int m0 = (blockIdx.y * 2 + wave) * 16;
  if (m0 >= Mrows) return;
  const int lo = lane & 15;   // row (A) / col (B,C)
  const int hi = lane >> 4;   // K-pair select

  const float* arow = A + (size_t)(m0 + lo) * K + 2 * hi;
  const float* wcol = W + (size_t)hi * N + n0 + lo;

  v8f acc0 = {}, acc1 = {}, acc2 = {}, acc3 = {};
  for (int k = 0; k < K; k += 4) {
    v2f a;
    a.x = arow[k];
    a.y = arow[k + 1];
    const float* wk0 = wcol + (size_t)k * N;
    const float* wk2 = wcol + (size_t)(k + 2) * N;
    v2f b0; b0.x = wk0[0];  b0.y = wk2[0];
    v2f b1; b1.x = wk0[16]; b1.y = wk2[16];
    v2f b2; b2.x = wk0[32]; b2.y = wk2[32];
    v2f b3; b3.x = wk0[48]; b3.y = wk2[48];
    acc0 = __builtin_amdgcn_wmma_f32_16x16x4_f32(false, a, false, b0, (short)0, acc0, false, false);
    acc1 = __builtin_amdgcn_wmma_f32_16x16x4_f32(false, a, false, b1, (short)0, acc1, false, false);
    acc2 = __builtin_amdgcn_wmma_f32_16x16x4_f32(false, a, false, b2, (short)0, acc2, false, false);
    acc3 = __builtin_amdgcn_wmma_f32_16x16x4_f32(false, a, false, b3, (short)0, acc3, false, false);
  }
  // D layout: VGPR j -> row m0 + j + 8*hi, col n0 + j16 + lo
  float* orow = Out + (size_t)(m0 + 8 * hi) * N + n0 + lo;
  const float bb0 = bias[n0 + lo];
  const float bb1 = bias[n0 + 16 + lo];
  const float bb2 = bias[n0 + 32 + lo];
  const float bb3 = bias[n0 + 48 + lo];
#pragma unroll
  for (int j = 0; j < 8; ++j) {
    float* p = orow + (size_t)j * N;
    p[0]  = acc0[j] + bb0;
    p[16] = acc1[j] + bb1;
    p[32] = acc2[j] + bb2;
    p[48] = acc3[j] + bb3;
  }
}

// ---------------------------------------------------------------------------
// span = mean(mention_text_feature[b, start..end, :])   -> [B, D]
// ---------------------------------------------------------------------------
__global__ void span_mean_k(const float4* __restrict__ feat, const int* __restrict__ start,
                            const int* __restrict__ end, float4* __restrict__ out) {
  const int b = blockIdx.x;
  const int s = start[b], e = end[b];
  const float inv = 1.0f / fmaxf((float)(e - s + 1), 1e-9f);
  for (int d = threadIdx.x; d < D4_; d += blockDim.x) {
    float4 acc = make_float4(0.f, 0.f, 0.f, 0.f);
    for (int t = s; t <= e; ++t) {
      const float4 f = feat[((size_t)b * LM_ + t) * D4_ + d];
      acc.x += f.x; acc.y += f.y; acc.z += f.z; acc.w += f.w;
    }
    out[(size_t)b * D4_ + d] = make_float4(acc.x * inv, acc.y * inv, acc.z * inv, acc.w * inv);
  }
}

// ent_rep = masked mean over LE of entity_text_feature  -> [B*C, D]
__global__ void ent_mean_k(const float4* __restrict__ feat, const int* __restrict__ mask,
                           float4* __restrict__ out) {
  const int bc = blockIdx.x;
  const float4* f = feat + (size_t)bc * LE_ * D4_;
  const int* mk = mask + (size_t)bc * LE_;
  float cnt = 0.f;
  for (int l = 0; l < LE_; ++l) cnt += (float)mk[l];
  const float inv = 1.0f / fmaxf(cnt, 1e-9f);
  for (int d = threadIdx.x; d < D4_; d += blockDim.x) {
    float4 acc = make_float4(0.f, 0.f, 0.f, 0.f);
    for (int l = 0; l < LE_; ++l) {
      const float m = (float)mk[l];
      const float4 fv = f[(size_t)l * D4_ + d];
      acc.x += fv.x * m; acc.y += fv.y * m; acc.z += fv.z * m; acc.w += fv.w * m;
    }
    out[(size_t)bc * D4_ + d] = make_float4(acc.x * inv, acc.y * inv, acc.z * inv, acc.w * inv);
  }
}

// out[row, r4] = mean over mid of in[row, mid, r4]  (b128 loads, coalesced)
__global__ void mean_mid4_k(const float4* __restrict__ in, float4* __restrict__ out,
                            int rows, int mid, int inner4) {
  const float inv = 1.0f / (float)mid;
  const long long total = (long long)rows * inner4;
  for (long long i = blockIdx.x * (long long)blockDim.x + threadIdx.x; i < total;
       i += (long long)gridDim.x * blockDim.x) {
    const long long row = i / inner4;
    const long long r = i - row * inner4;
    const float4* p = in + (row * (long long)mid) * inner4 + r;
    float4 acc = make_float4(0.f, 0.f, 0.f, 0.f);
    for (int m = 0; m < mid; ++m) {
      const float4 f = p[(long long)m * inner4];
      acc.x += f.x; acc.y += f.y; acc.z += f.z; acc.w += f.w;
    }
    out[i] = make_float4(acc.x * inv, acc.y * inv, acc.z * inv, acc.w * inv);
  }
}

// e0[b,c] = cos(span[b], entity_text_feature[b,c,0,:])
__global__ void mtet_k(const float4* __restrict__ span, const float4* __restrict__ etf,
                       float* __restrict__ e0) {
  __shared__ float sm[256];
  const int bc = blockIdx.x;
  const int b = bc / C_;
  const float4* x = span + (size_t)b * D4_;
  const float4* y = etf + (size_t)bc * LE_ * D4_;  // token 0
  float dot = 0.f, nx = 0.f, ny = 0.f;
  for (int d = threadIdx.x; d < D4_; d += 256) {
    const float4 xv = x[d], yv = y[d];
    dot += dot4(xv, yv); nx += dot4(xv, xv); ny += dot4(yv, yv);
  }
  dot = block_sum(dot, sm);
  nx = block_sum(nx, sm);
  ny = block_sum(ny, sm);
  if (threadIdx.x == 0) e0[bc] = dot / fmaxf(sqrtf(nx) * sqrtf(ny), 1e-8f);
}

__global__ void scale_edges_k(const float* __restrict__ mtei, const float* __restrict__ miet,
                              float* __restrict__ e1, float* __restrict__ e2) {
  const int i = blockIdx.x * blockDim.x + threadIdx.x;
  if (i < B_ * C_) { e1[i] = mtei[i] * 0.01f; e2[i] = miet[i] * 0.01f; }
}

// e3[b,c] = score-weighted mean of pairwise object cosines
__global__ void miei_k(const float4* __restrict__ mo, const float* __restrict__ ms,
                       const float4* __restrict__ eo, const float* __restrict__ es,
                       float* __restrict__ e3) {
  __shared__ float sm[256];
  const int bc = blockIdx.x;
  const int b = bc / C_;
  const float4* mop = mo + (size_t)b * M_ * R4_;
  const float4* eop = eo + (size_t)bc * M_ * R4_;
  float dots[M_][M_] = {};
  float nm2[M_] = {}, ne2[M_] = {};
  for (int r = threadIdx.x; r < R4_; r += 256) {
    float4 mv[M_], ev[M_];
#pragma unroll
    for (int m = 0; m < M_; ++m) mv[m] = mop[(size_t)m * R4_ + r];
#pragma unroll
    for (int n = 0; n < M_; ++n) ev[n] = eop[(size_t)n * R4_ + r];
#pragma unroll
    for (int m = 0; m < M_; ++m) {
      nm2[m] += dot4(mv[m], mv[m]);
#pragma unroll
      for (int n = 0; n < M_; ++n) dots[m][n] += dot4(mv[m], ev[n]);
    }
#pragma unroll
    for (int n = 0; n < M_; ++n) ne2[n] += dot4(ev[n], ev[n]);
  }
  float dred[M_][M_], nmr[M_], ner[M_];
  for (int m = 0; m < M_; ++m)
    for (int n = 0; n < M_; ++n) dred[m][n] = block_sum(dots[m][n], sm);
  for (int m = 0; m < M_; ++m) nmr[m] = block_sum(nm2[m], sm);
  for (int n = 0; n < M_; ++n) ner[n] = block_sum(ne2[n], sm);
  if (threadIdx.x == 0) {
    float num = 0.f, den = 0.f;
    for (int m = 0; m < M_; ++m) {
      const float sc_m = ms[(size_t)b * M_ + m];
      const float nmm = sqrtf(nmr[m]);
      for (int n = 0; n < M_; ++n) {
        const float sim = dred[m][n] / fmaxf(nmm * sqrtf(ner[n]), 1e-8f);
        const float sc = sc_m * es[(size_t)bc * M_ + n];
        num += sim * sc;
        den += sc;
      }
    }
    e3[bc] = num / (den + 1e-9f);
  }
}

// mention-vertex aggregation: s0 = v0 + mean_c(e0*v2 + e1*v3); s1 analogous
__global__ void agg_mention_k(const float4* __restrict__ v0, const float4* __restrict__ v1,
                              const float4* __restrict__ v2, const float4* __restrict__ v3,
                              const float* __restrict__ e0, const float* __restrict__ e1,
                              const float* __restrict__ e2, const float* __restrict__ e3,
                              float4* __restrict__ s0, float4* __restrict__ s1) {
  const int i = blockIdx.x * blockDim.x + threadIdx.x;
  if (i >= B_ * D4_) return;
  const int b = i / D4_;
  const int d = i - b * D4_;
  float4 a0 = make_float4(0.f, 0.f, 0.f, 0.f);
  float4 a1 = make_float4(0.f, 0.f, 0.f, 0.f);
  for (int c = 0; c < C_; ++c) {
    const int bc = b * C_ + c;
    const float4 x2 = v2[(size_t)bc * D4_ + d];
    const float4 x3 = v3[(size_t)bc * D4_ + d];
    const float w0 = e0[bc], w1 = e1[bc], w2 = e2[bc], w3 = e3[bc];
    a0.x += w0 * x2.x + w1 * x3.x;  a0.y += w0 * x2.y + w1 * x3.y;
    a0.z += w0 * x2.z + w1 * x3.z;  a0.w += w0 * x2.w + w1 * x3.w;
    a1.x += w2 * x2.x + w3 * x3.x;  a1.y += w2 * x2.y + w3 * x3.y;
    a1.z += w2 * x2.z + w3 * x3.z;  a1.w += w2 * x2.w + w3 * x3.w;
  }
  const float invC = 1.0f / (float)C_;
  const float4 b0 = v0[i], b1 = v1[i];
  s0[i] = make_float4(b0.x + a0.x * invC, b0.y + a0.y * invC,
                      b0.z + a0.z * invC, b0.w + a0.w * invC);
  s1[i] = make_float4(b1.x + a1.x * invC, b1.y + a1.y * invC,
                      b1.z + a1.z * invC, b1.w + a1.w * invC);
}

// entity-vertex aggregation: s2 = v2 + e0*v0 + e2*v1; s3 = v3 + e1*v0 + e3*v1
__global__ void agg_entity_k(const float4* __restrict__ v0, const float4* __restrict__ v1,
                             const float4* __restrict__ v2, const float4* __restrict__ v3,
                             const float* __restrict__ e0, const float* __restrict__ e1,
                             const float* __restrict__ e2, const float* __restrict__ e3,
                             float4* __restrict__ s2, float4* __restrict__ s3) {
  const int i = blockIdx.x * blockDim.x + threadIdx.x;
  if (i >= B_ * C_ * D4_) return;
  const int bc = i / D4_;
  const int d = i - bc * D4_;
  const int b = bc / C_;
  const float4 x0 = v0[(size_t)b * D4_ + d];
  const float4 x1 = v1[(size_t)b * D4_ + d];
  const float4 y2 = v2[i], y3 = v3[i];
  const float w0 = e0[bc], w1 = e1[bc], w2 = e2[bc], w3 = e3[bc];
  s2[i] = make_float4(y2.x + w0 * x0.x + w2 * x1.x, y2.y + w0 * x0.y + w2 * x1.y,
                      y2.z + w0 * x0.z + w2 * x1.z, y2.w + w0 * x0.w + w2 * x1.w);
  s3[i] = make_float4(y3.x + w1 * x0.x + w3 * x1.x, y3.y + w1 * x0.y + w3 * x1.y,
                      y3.z + w1 * x0.z + w3 * x1.z, y3.w + w1 * x0.w + w3 * x1.w);
}

// in-place LayerNorm + ReLU per row of D_
__global__ void ln_relu_k(float4* __restrict__ x, const float4* __restrict__ g,
                          const float4* __restrict__ bt) {
  __shared__ float sm[256];
  float4* p = x + (size_t)blockIdx.x * D4_;
  float s1 = 0.f, s2 = 0.f;
  for (int d = threadIdx.x; d < D4_; d += 256) {
    const float4 v = p[d];
    s1 += v.x + v.y + v.z + v.w;
    s2 += dot4(v, v);
  }
  s1 = block_sum(s1, sm);
  s2 = block_sum(s2, sm);
  const float mu = s1 / (float)D_;
  const float var = s2 / (float)D_ - mu * mu;
  const float rstd = rsqrtf(var + 1e-5f);
  for (int d = threadIdx.x; d < D4_; d += 256) {
    const float4 v = p[d];
    const float4 gg = g[d], bb = bt[d];
    p[d] = make_float4(fmaxf((v.x - mu) * rstd * gg.x + bb.x, 0.f),
                       fmaxf((v.y - mu) * rstd * gg.y + bb.y, 0.f),
                       fmaxf((v.z - mu) * rstd * gg.z + bb.z, 0.f),
                       fmaxf((v.w - mu) * rstd * gg.w + bb.w, 0.f));
  }
}

// new_e[j] = relu(mean_d(fu[ui]*fv[vi]) + e[j]) for the 4 fixed edges
__global__ void edge_update_k(const float4* __restrict__ fu0, const float4* __restrict__ fu1,
                              const float4* __restrict__ fv2, const float4* __restrict__ fv3,
                              const float* __restrict__ e0, const float* __restrict__ e1,
                              const float* __restrict__ e2, const float* __restrict__ e3,
                              float* __restrict__ n0, float* __restrict__ n1,
                              float* __restrict__ n2, float* __restrict__ n3) {
  __shared__ float sm[256];
  const int bc = blockIdx.x;
  const int b = bc / C_;
  const float4* u0 = fu0 + (size_t)b * D4_;
  const float4* u1 = fu1 + (size_t)b * D4_;
  const float4* w2 = fv2 + (size_t)bc * D4_;
  const float4* w3 = fv3 + (size_t)bc * D4_;
  float a02 = 0.f, a03 = 0.f, a12 = 0.f, a13 = 0.f;
  for (int d = threadIdx.x; d < D4_; d += 256) {
    const float4 x0 = u0[d], x1 = u1[d], y2 = w2[d], y3 = w3[d];
    a02 += dot4(x0, y2); a03 += dot4(x0, y3);
    a12 += dot4(x1, y2); a13 += dot4(x1, y3);
  }
  a02 = block_sum(a02, sm);
  a03 = block_sum(a03, sm);
  a12 = block_sum(a12, sm);
  a13 = block_sum(a13, sm);
  if (threadIdx.x == 0) {
    const float invD = 1.0f / (float)D_;
    n0[bc] = fmaxf(a02 * invD + e0[bc], 0.f);
    n1[bc] = fmaxf(a03 * invD + e1[bc], 0.f);
    n2[bc] = fmaxf(a12 * invD + e2[bc], 0.f);
    n3[bc] = fmaxf(a13 * invD + e3[bc], 0.f);
  }
}

// out[b,c] = cos(v0[b], v2[b,c])
__global__ void cos_out_k(const float4* __restrict__ v0, const float4* __restrict__ v2,
                          float* __restrict__ out) {
  __shared__ float sm[256];
  const int bc = blockIdx.x;
  const int b = bc / C_;
  const float4* x = v0 + (size_t)b * D4_;
  const float4* y = v2 + (size_t)bc * D4_;
  float dot = 0.f, nx = 0.f, ny = 0.f;
  for (int d = threadIdx.x; d < D4_; d += 256) {
    const float4 xv = x[d], yv = y[d];
    dot += dot4(xv, yv); nx += dot4(xv, xv); ny += dot4(yv, yv);
  }
  dot = block_sum(dot, sm);
  nx = block_sum(nx, sm);
  ny = block_sum(ny, sm);
  if (threadIdx.x == 0) out[bc] = dot / fmaxf(sqrtf(nx) * sqrtf(ny), 1e-8f);
}

// ---------------------------------------------------------------------------
extern "C" void kernel_launch(void* const* d_in, const int* in_sizes, int n_in,
                              void* d_out, int out_size, void* d_ws, size_t ws_size,
                              hipStream_t stream) {
  (void)in_sizes; (void)n_in; (void)out_size; (void)ws_size;
  const float* mtf    = (const float*)d_in[0];
  const int*   mstart = (const int*)d_in[2];
  const int*   mend   = (const int*)d_in[3];
  const float* mif    = (const float*)d_in[4];
  const float* mof    = (const float*)d_in[5];
  const float* mos    = (const float*)d_in[6];
  const float* etf    = (const float*)d_in[7];
  const int*   etm    = (const int*)d_in[8];
  const float* eif    = (const float*)d_in[9];
  const float* eof    = (const float*)d_in[10];
  const float* eos    = (const float*)d_in[11];
  const float* miet   = (const float*)d_in[12];
  const float* mtei   = (const float*)d_in[13];
  const float* w_mt = (const float*)d_in[14]; const float* b_mt = (const float*)d_in[15];
  const float* w_et = (const float*)d_in[16]; const float* b_et = (const float*)d_in[17];
  const float* w_mi = (const float*)d_in[18]; const float* b_mi = (const float*)d_in[19];
  const float* w_ei = (const float*)d_in[20]; const float* b_ei = (const float*)d_in[21];
  const float* wh_w = (const float*)d_in[22]; const float* wh_b = (const float*)d_in[23];
  const float* wu_w = (const float*)d_in[24]; const float* wu_b = (const float*)d_in[25];
  const float* wv_w = (const float*)d_in[26]; const float* wv_b = (const float*)d_in[27];
  const float* ln_g = (const float*)d_in[28]; const float* ln_b = (const float*)d_in[29];
  float* out = (float*)d_out;

  // workspace carve-up (all f32; ~32.3 MB)
  float* wsp = (float*)d_ws;
  size_t off = 0;
  auto alloc = [&](size_t n) { float* p = wsp + off; off += n; return p; };
  const size_t BD = (size_t)B_ * D_, BCD = (size_t)B_ * C_ * D_, BC = (size_t)B_ * C_;
  float* span   = alloc(BD);
  float* erep   = alloc(BCD);
  float* mimean = alloc((size_t)B_ * R_);
  float* eimean = alloc((size_t)B_ * C_ * R_);
  float* v[4]  = { alloc(BD), alloc(BD), alloc(BCD), alloc(BCD) };
  float* nv[4] = { alloc(BD), alloc(BD), alloc(BCD), alloc(BCD) };
  float* fu0 = alloc(BD);  float* fu1 = alloc(BD);
  float* fv2 = alloc(BCD); float* fv3 = alloc(BCD);
  float* e[4]  = { alloc(BC), alloc(BC), alloc(BC), alloc(BC) };
  float* ne[4] = { alloc(BC), alloc(BC), alloc(BC), alloc(BC) };

  // ---- Vertex encoder reductions (HBM-bound, single pass, b128 loads) ----
  span_mean_k<<<B_, 256, 0, stream>>>((const float4*)mtf, mstart, mend, (float4*)span);
  ent_mean_k<<<B_ * C_, 256, 0, stream>>>((const float4*)etf, etm, (float4*)erep);
  mean_mid4_k<<<128, 256, 0, stream>>>((const float4*)mif, (float4*)mimean, B_, P_, R4_);
  mean_mid4_k<<<2048, 256, 0, stream>>>((const float4*)eif, (float4*)eimean, B_ * C_, P_, R4_);

  // ---- Vertex projections (WMMA GEMMs, N=768) ----
  gemm16_wmma<<<dim3(D_ / 64, 2), 64, 0, stream>>>(span, w_mt, b_mt, v[0], B_, D_, D_);
  gemm16_wmma<<<dim3(D_ / 64, 2), 64, 0, stream>>>(mimean, w_mi, b_mi, v[1], B_, R_, D_);
  gemm16_wmma<<<dim3(D_ / 64, 32), 64, 0, stream>>>(erep, w_et, b_et, v[2], B_ * C_, D_, D_);
  gemm16_wmma<<<dim3(D_ / 64, 32), 64, 0, stream>>>(eimean, w_ei, b_ei, v[3], B_ * C_, R_, D_);

  // ---- Edge encoder ----
  mtet_k<<<B_ * C_, 256, 0, stream>>>((const float4*)span, (const float4*)etf, e[0]);
  scale_edges_k<<<(B_ * C_ + 255) / 256, 256, 0, stream>>>(mtei, miet, e[1], e[2]);
  miei_k<<<B_ * C_, 256, 0, stream>>>((const float4*)mof, mos, (const float4*)eof, eos, e[3]);

  // ---- GCN layers ----
  for (int l = 0; l < NL_; ++l) {
    const float* wh = wh_w + (size_t)l * D_ * D_; const float* bh = wh_b + (size_t)l * D_;
    const float* wu = wu_w + (size_t)l * D_ * D_; const float* bu = wu_b + (size_t)l * D_;
    const float* wv = wv_w + (size_t)l * D_ * D_; const float* bv = wv_b + (size_t)l * D_;
    const float* g  = ln_g + (size_t)l * D_;      const float* bl = ln_b + (size_t)l * D_;

    // aggregate (into fu/fv as scratch), project, LN+ReLU
    agg_mention_k<<<(int)((BD / 4 + 255) / 256), 256, 0, stream>>>(
        (const float4*)v[0], (const float4*)v[1], (const float4*)v[2], (const float4*)v[3],
        e[0], e[1], e[2], e[3], (float4*)fu0, (float4*)fu1);
    agg_entity_k<<<(int)((BCD / 4 + 255) / 256), 256, 0, stream>>>(
        (const float4*)v[0], (const float4*)v[1], (const float4*)v[2], (const float4*)v[3],
        e[0], e[1], e[2], e[3], (float4*)fv2, (float4*)fv3);
    gemm16_wmma<<<dim3(D_ / 64, 2), 64, 0, stream>>>(fu0, wh, bh, nv[0], B_, D_, D_);
    gemm16_wmma<<<dim3(D_ / 64, 2), 64, 0, stream>>>(fu1, wh, bh, nv[1], B_, D_, D_);
    gemm16_wmma<<<dim3(D_ / 64, 32), 64, 0, stream>>>(fv2, wh, bh, nv[2], B_ * C_, D_, D_);
    gemm16_wmma<<<dim3(D_ / 64, 32), 64, 0, stream>>>(fv3, wh, bh, nv[3], B_ * C_, D_, D_);
    ln_relu_k<<<B_, 256, 0, stream>>>((float4*)nv[0], (const float4*)g, (const float4*)bl);
    ln_relu_k<<<B_, 256, 0, stream>>>((float4*)nv[1], (const float4*)g, (const float4*)bl);
    ln_relu_k<<<B_ * C_, 256, 0, stream>>>((float4*)nv[2], (const float4*)g, (const float4*)bl);
    ln_relu_k<<<B_ * C_, 256, 0, stream>>>((float4*)nv[3], (const float4*)g, (const float4*)bl);

    // edge update uses OLD vertexes / OLD edges
    gemm16_wmma<<<dim3(D_ / 64, 2), 64, 0, stream>>>(v[0], wu, bu, fu0, B_, D_, D_);
    gemm16_wmma<<<dim3(D_ / 64, 2), 64, 0, stream>>>(v[1], wu, bu, fu1, B_, D_, D_);
    gemm16_wmma<<<dim3(D_ / 64, 32), 64, 0, stream>>>(v[2], wv, bv, fv2, B_ * C_, D_, D_);
    gemm16_wmma<<<dim3(D_ / 64, 32), 64, 0, stream>>>(v[3], wv, bv, fv3, B_ * C_, D_, D_);
    edge_update_k<<<B_ * C_, 256, 0, stream>>>(
        (const float4*)fu0, (const float4*)fu1, (const float4*)fv2, (const float4*)fv3,
        e[0], e[1], e[2], e[3], ne[0], ne[1], ne[2], ne[3]);

    for (int j = 0; j < 4; ++j) { float* t = v[j]; v[j] = nv[j]; nv[j] = t; }
    for (int j = 0; j < 4; ++j) { float* t = e[j]; e[j] = ne[j]; ne[j] = t; }
  }

  // ---- output cosine ----
  cos_out_k<<<B_ * C_, 256, 0, stream>>>((const float4*)v[0], (const float4*)v[2], out);
}